// RoIPooling2D_51883204935936
// MI455X (gfx1250) — compile-verified
//
#include <hip/hip_runtime.h>

// RoI max pooling, MI455X (gfx1250, wave32).
// x: (1, 512, 50, 50) f32, rois: (1024, 5) f32, out: (1024, 512, 7, 7) f32.
// Store-bandwidth bound: one thread computes 4 consecutive flat outputs and
// issues a single 128-bit non-temporal store. The 4 elements provably share
// the same RoI (C*7*7 = 25088 is divisible by 4), so the RoI load, the
// round-half-even coordinate snapping, and the bin-size divides are hoisted
// out of the per-element loop. x (5.12 MB) is L2-resident; window reads hit
// cache, writes stream to HBM with NT hint.

#define C_    512
#define H_    50
#define W_    50
#define OUTH_ 7
#define OUTW_ 7
#define SCALE_ 0.0625f

typedef float v4f __attribute__((ext_vector_type(4)));

__device__ __forceinline__ float bin_max(const float* __restrict__ plane,
                                         int hstart, int hend,
                                         int wstart, int wend) {
    float best = -__builtin_inff();
    for (int h = hstart; h < hend; ++h) {
        const float* row = plane + h * W_ + wstart;
        int nw = wend - wstart;
        for (int w = 0; w < nw; ++w) {
            float v = row[w];
            best = (v > best) ? v : best;
        }
    }
    // Empty bin (or all -inf) -> 0, matching the reference's bi<0 path.
    return (best == -__builtin_inff()) ? 0.0f : best;
}

__global__ __launch_bounds__(256) void roi_pool_vec4_kernel(
        const float* __restrict__ x, const float* __restrict__ rois,
        float* __restrict__ out, int nvec) {
    int t = blockIdx.x * blockDim.x + threadIdx.x;
    if (t >= nvec) return;
    int e = t * 4;

    // All 4 elements share the same RoI row n (25088 % 4 == 0).
    int n = e / (C_ * OUTH_ * OUTW_);
    int r = e - n * (C_ * OUTH_ * OUTW_);

    const float* roi = rois + n * 5;
    __builtin_prefetch(roi, 0, 1);   // gfx1250 global_prefetch_b8

    int b  = (int)roi[0];
    // jnp.round == round-half-to-even == rintf (v_rndne_f32)
    int ws = (int)rintf(roi[1] * SCALE_);
    int hs = (int)rintf(roi[2] * SCALE_);
    int we = (int)rintf(roi[3] * SCALE_);
    int he = (int)rintf(roi[4] * SCALE_);

    float bin_w = (float)max(we - ws + 1, 1) / (float)OUTW_;  // IEEE f32 divide
    float bin_h = (float)max(he - hs + 1, 1) / (float)OUTH_;

    const float* xb = x + (size_t)b * (C_ * H_ * W_);

    v4f o;
#pragma unroll
    for (int k = 0; k < 4; ++k) {
        int rk = r + k;
        int c  = rk / (OUTH_ * OUTW_);
        int s  = rk - c * (OUTH_ * OUTW_);
        int ph = s / OUTW_;
        int pw = s - ph * OUTW_;

        int hstart = min(max((int)floorf(bin_h * (float)ph)       + hs, 0), H_);
        int hend   = min(max((int)ceilf (bin_h * (float)(ph + 1)) + hs, 0), H_);
        int wstart = min(max((int)floorf(bin_w * (float)pw)       + ws, 0), W_);
        int wend   = min(max((int)ceilf (bin_w * (float)(pw + 1)) + ws, 0), W_);

        o[k] = bin_max(xb + (size_t)c * (H_ * W_), hstart, hend, wstart, wend);
    }

    // Streaming write-once output: non-temporal 128-bit store.
    __builtin_nontemporal_store(o, (v4f*)out + t);
}

// Defensive scalar tail (out_size here is divisible by 4, so normally unused).
__global__ __launch_bounds__(64) void roi_pool_tail_kernel(
        const float* __restrict__ x, const float* __restrict__ rois,
        float* __restrict__ out, int start, int total) {
    int e = start + blockIdx.x * blockDim.x + threadIdx.x;
    if (e >= total) return;

    int n  = e / (C_ * OUTH_ * OUTW_);
    int r  = e - n * (C_ * OUTH_ * OUTW_);
    int c  = r / (OUTH_ * OUTW_);
    int s  = r - c * (OUTH_ * OUTW_);
    int ph = s / OUTW_;
    int pw = s - ph * OUTW_;

    const float* roi = rois + n * 5;
    int b  = (int)roi[0];
    int ws = (int)rintf(roi[1] * SCALE_);
    int hs = (int)rintf(roi[2] * SCALE_);
    int we = (int)rintf(roi[3] * SCALE_);
    int he = (int)rintf(roi[4] * SCALE_);

    float bin_w = (float)max(we - ws + 1, 1) / (float)OUTW_;
    float bin_h = (float)max(he - hs + 1, 1) / (float)OUTH_;

    int hstart = min(max((int)floorf(bin_h * (float)ph)       + hs, 0), H_);
    int hend   = min(max((int)ceilf (bin_h * (float)(ph + 1)) + hs, 0), H_);
    int wstart = min(max((int)floorf(bin_w * (float)pw)       + ws, 0), W_);
    int wend   = min(max((int)ceilf (bin_w * (float)(pw + 1)) + ws, 0), W_);

    out[e] = bin_max(x + (size_t)(b * C_ + c) * (H_ * W_),
                     hstart, hend, wstart, wend);
}

extern "C" void kernel_launch(void* const* d_in, const int* in_sizes, int n_in,
                              void* d_out, int out_size, void* d_ws, size_t ws_size,
                              hipStream_t stream) {
    const float* x    = (const float*)d_in[0];  // (1,512,50,50)
    const float* rois = (const float*)d_in[1];  // (1024,5)
    float* out = (float*)d_out;                 // (1024,512,7,7) flat

    int nvec = out_size >> 2;
    int rem  = out_size - (nvec << 2);

    if (nvec > 0) {
        int blocks = (nvec + 255) / 256;
        roi_pool_vec4_kernel<<<blocks, 256, 0, stream>>>(x, rois, out, nvec);
    }
    if (rem > 0) {
        roi_pool_tail_kernel<<<1, 64, 0, stream>>>(x, rois, out, nvec << 2, out_size);
    }
}